// AggregationLayer_62775241999269
// MI455X (gfx1250) — compile-verified
//
#include <hip/hip_runtime.h>
#include <math.h>

typedef float v2f __attribute__((ext_vector_type(2)));
typedef float v8f __attribute__((ext_vector_type(8)));

static constexpr int H  = 768;     // hidden dim
static constexpr int NR = 32768;   // rows of fact
static constexpr int MR = 4096;    // rows of elements_p
static constexpr float NEG_BIG = -3.402823466e+38f;

// ---------------------------------------------------------------------------
// Wave-level batched dot product via fp32 WMMA (V_WMMA_F32_16X16X4_F32).
// Computes dot(src[row0+m], tvec) for m = 0..15 for one wave32.
// A layout (32-bit 16x4, ISA 7.12.2): lane L in [0,16): row M=L, VGPR0=K0,
// VGPR1=K1; lanes [16,32): same rows, K2/K3.  B is broadcast across all 16
// columns (B[k][n] = tvec[k]) so every D column equals the dot vector:
// half-0 lanes hold m=0..7 in acc[0..7], half-1 lanes hold m=8..15.
// Requires EXEC all ones (callers use full 256-thread blocks, no divergence).
// ---------------------------------------------------------------------------
__device__ __forceinline__ v8f wmma_dot16(const float* __restrict__ src,
                                          int row0,
                                          const float* __restrict__ tvec) {
  const int lane = threadIdx.x & 31;
  const int half = lane >> 4;       // 0: K0/K1, 1: K2/K3
  const int mrow = lane & 15;
  const float* base = src + (size_t)(row0 + mrow) * H + half * 2;
  v8f acc = {0.f, 0.f, 0.f, 0.f, 0.f, 0.f, 0.f, 0.f};
  for (int k = 0; k < H; k += 4) {
    v2f a; a.x = base[k];                 a.y = base[k + 1];
    v2f b; b.x = tvec[k + half * 2];      b.y = tvec[k + half * 2 + 1];
    acc = __builtin_amdgcn_wmma_f32_16x16x4_f32(
        /*neg_a=*/false, a, /*neg_b=*/false, b,
        /*c_mod=*/(short)0, acc, /*reuse_a=*/false, /*reuse_b=*/false);
  }
  return acc;
}

// K1: per-row-chunk column max of fact. grid (3,128), block 256.
__global__ void k1_colmax_part(const float* __restrict__ fact, float* __restrict__ part) {
  const int col = blockIdx.x * 256 + threadIdx.x;
  const float* base = fact + (size_t)blockIdx.y * 256 * H + col;
  float m = NEG_BIG;
  for (int r = 0; r < 256; ++r) m = fmaxf(m, base[(size_t)r * H]);
  part[(size_t)blockIdx.y * H + col] = m;
}

// K2: reduce 128 partials -> fact_Q. grid 3, block 256.
__global__ void k2_colmax_red(const float* __restrict__ part, float* __restrict__ fq) {
  const int col = blockIdx.x * 256 + threadIdx.x;
  float m = NEG_BIG;
  for (int b = 0; b < 128; ++b) m = fmaxf(m, part[(size_t)b * H + col]);
  fq[col] = m;
}

// K3: q = fact_Q @ W  (q[j] = sum_h fq[h]*W[h*H+j]).  grid 3, block 256.
__global__ void k3_q(const float* __restrict__ fq, const float* __restrict__ W,
                     float* __restrict__ q) {
  __shared__ float s_fq[H];
  const int t = threadIdx.x;
  for (int i = t; i < H; i += 256) s_fq[i] = fq[i];
  __syncthreads();
  const int j = blockIdx.x * 256 + t;
  float v = 0.f;
  for (int h = 0; h < H; ++h) v = fmaf(s_fq[h], W[(size_t)h * H + j], v);
  q[j] = v;
}

// K4: f2e logits via WMMA: f2e[m] = q . ep[m].  grid 32, block 256 (8 waves x 16 rows).
__global__ void k4_f2e_logits(const float* __restrict__ ep, const float* __restrict__ q,
                              float* __restrict__ f2e) {
  __shared__ float s_q[H];
  const int t = threadIdx.x;
  for (int i = t; i < H; i += 256) s_q[i] = q[i];
  __syncthreads();
  const int row0 = blockIdx.x * 128 + (t >> 5) * 16;
  v8f d = wmma_dot16(ep, row0, s_q);
  const int lane = t & 31;
  if (lane == 0)  for (int r = 0; r < 8; ++r) f2e[row0 + r]     = d[r];
  if (lane == 16) for (int r = 0; r < 8; ++r) f2e[row0 + 8 + r] = d[r];
}

// K4b: in-place softmax over the 4096 logits. grid 1, block 256 (16 per thread).
__global__ void k4b_softmax(float* __restrict__ f2e) {
  __shared__ float s_red[256];
  const int t = threadIdx.x;
  float l[16];
  float m = NEG_BIG;
  for (int i = 0; i < 16; ++i) { l[i] = f2e[t * 16 + i]; m = fmaxf(m, l[i]); }
  s_red[t] = m; __syncthreads();
  for (int off = 128; off > 0; off >>= 1) {
    if (t < off) s_red[t] = fmaxf(s_red[t], s_red[t + off]);
    __syncthreads();
  }
  const float M = s_red[0]; __syncthreads();
  float s = 0.f;
  for (int i = 0; i < 16; ++i) s += expf(l[i] - M);
  s_red[t] = s; __syncthreads();
  for (int off = 128; off > 0; off >>= 1) {
    if (t < off) s_red[t] += s_red[t + off];
    __syncthreads();
  }
  const float inv = 1.f / s_red[0];
  for (int i = 0; i < 16; ++i) f2e[t * 16 + i] = expf(l[i] - M) * inv;
}

// K4c: partial weighted sums of elements_p. grid 16, block 256.
__global__ void k4c_ep_partial(const float* __restrict__ ep, const float* __restrict__ p,
                               float* __restrict__ part) {
  __shared__ float s_p[256];
  const int t = threadIdx.x;
  const int b = blockIdx.x;
  s_p[t] = p[b * 256 + t];
  __syncthreads();
  const float* base = ep + (size_t)b * 256 * H;
  float a0 = 0.f, a1 = 0.f, a2 = 0.f;
  for (int r = 0; r < 256; ++r) {
    const float w = s_p[r];
    const float* rp = base + (size_t)r * H;
    a0 = fmaf(w, rp[t],       a0);
    a1 = fmaf(w, rp[t + 256], a1);
    a2 = fmaf(w, rp[t + 512], a2);
  }
  part[(size_t)b * H + t]       = a0;
  part[(size_t)b * H + t + 256] = a1;
  part[(size_t)b * H + t + 512] = a2;
}

// K4d: combine 16 partials -> elements_p_; also write to d_out[H..2H).  grid 3.
__global__ void k4d_ep_combine(const float* __restrict__ part, float* __restrict__ ep_,
                               float* __restrict__ out) {
  const int col = blockIdx.x * 256 + threadIdx.x;
  float v = 0.f;
  for (int b = 0; b < 16; ++b) v += part[(size_t)b * H + col];
  ep_[col] = v;
  out[H + col] = v;
}

// K4e: w_v = W @ elements_p_ (w_v[i] = sum_j W[i*H+j]*ep_[j]). grid 3.
__global__ void k4e_wv(const float* __restrict__ W, const float* __restrict__ ep_,
                       float* __restrict__ wv) {
  __shared__ float s_e[H];
  const int t = threadIdx.x;
  for (int i = t; i < H; i += 256) s_e[i] = ep_[i];
  __syncthreads();
  const int i = blockIdx.x * 256 + t;
  const float* row = W + (size_t)i * H;
  float v = 0.f;
  for (int j = 0; j < H; ++j) v = fmaf(row[j], s_e[j], v);
  wv[i] = v;
}

// K5: fused e2f pass. Each block: 128 rows of fact. WMMA logits (8 waves x 16
// rows), block-local softmax partials (m, s), weighted row-sum acc[768].
// grid 256, block 256.
__global__ void k5_e2f_partial(const float* __restrict__ fact, const float* __restrict__ wv,
                               float* __restrict__ em, float* __restrict__ es,
                               float* __restrict__ eacc) {
  __shared__ float s_wv[H];
  __shared__ float s_logit[128];
  __shared__ float s_red[256];
  const int t = threadIdx.x;
  for (int i = t; i < H; i += 256) s_wv[i] = wv[i];
  __syncthreads();

  const int b = blockIdx.x;
  const int row_base = b * 128;
  const int row0 = row_base + (t >> 5) * 16;
  v8f d = wmma_dot16(fact, row0, s_wv);
  const int lane = t & 31;
  const int w16 = (t >> 5) * 16;
  if (lane == 0)  for (int r = 0; r < 8; ++r) s_logit[w16 + r]     = d[r];
  if (lane == 16) for (int r = 0; r < 8; ++r) s_logit[w16 + 8 + r] = d[r];
  __syncthreads();

  // block max over 128 logits
  s_red[t] = (t < 128) ? s_logit[t] : NEG_BIG;
  __syncthreads();
  for (int off = 128; off > 0; off >>= 1) {
    if (t < off) s_red[t] = fmaxf(s_red[t], s_red[t + off]);
    __syncthreads();
  }
  const float m = s_red[0]; __syncthreads();

  // unnormalized probs + block sum
  float pe = 0.f;
  if (t < 128) { pe = expf(s_logit[t] - m); s_logit[t] = pe; }
  s_red[t] = pe; __syncthreads();
  for (int off = 128; off > 0; off >>= 1) {
    if (t < off) s_red[t] += s_red[t + off];
    __syncthreads();
  }
  const float s = s_red[0]; __syncthreads();

  // weighted row-sum: re-read this block's fact tile (hot in cache)
  const float* fb = fact + (size_t)row_base * H;
  float a0 = 0.f, a1 = 0.f, a2 = 0.f;
  for (int r = 0; r < 128; ++r) {
    const float w = s_logit[r];
    const float* rp = fb + (size_t)r * H;
    __builtin_prefetch(rp + 4 * (size_t)H, 0, 0);
    a0 = fmaf(w, rp[t],       a0);
    a1 = fmaf(w, rp[t + 256], a1);
    a2 = fmaf(w, rp[t + 512], a2);
  }
  eacc[(size_t)b * H + t]       = a0;
  eacc[(size_t)b * H + t + 256] = a1;
  eacc[(size_t)b * H + t + 512] = a2;
  if (t == 0) { em[b] = m; es[b] = s; }
}

// K6: combine 256 block partials -> fact_ in d_out[0..H). grid 3, block 256.
__global__ void k6_combine(const float* __restrict__ em, const float* __restrict__ es,
                           const float* __restrict__ eacc, float* __restrict__ out) {
  __shared__ float s_m[256];
  __shared__ float s_red[256];
  const int t = threadIdx.x;
  const float mt = em[t];
  s_m[t] = mt;
  s_red[t] = mt;
  __syncthreads();
  for (int off = 128; off > 0; off >>= 1) {
    if (t < off) s_red[t] = fmaxf(s_red[t], s_red[t + off]);
    __syncthreads();
  }
  const float M = s_red[0]; __syncthreads();
  s_red[t] = es[t] * expf(mt - M);
  __syncthreads();
  for (int off = 128; off > 0; off >>= 1) {
    if (t < off) s_red[t] += s_red[t + off];
    __syncthreads();
  }
  const float Z = s_red[0]; __syncthreads();
  s_m[t] = expf(mt - M);
  __syncthreads();
  const int col = blockIdx.x * 256 + t;
  float v = 0.f;
  for (int b = 0; b < 256; ++b) v = fmaf(eacc[(size_t)b * H + col], s_m[b], v);
  out[col] = v / Z;
}

extern "C" void kernel_launch(void* const* d_in, const int* in_sizes, int n_in,
                              void* d_out, int out_size, void* d_ws, size_t ws_size,
                              hipStream_t stream) {
  const float* fact = (const float*)d_in[0];   // [32768,768]
  const float* ep   = (const float*)d_in[1];   // [4096,768]
  const float* W    = (const float*)d_in[2];   // [768,768]
  float* out = (float*)d_out;                  // [768 fact_] ++ [768 elements_p_]

  // workspace layout (floats)
  float* ws      = (float*)d_ws;
  float* w_cmax  = ws;                    // 128*H
  float* w_fq    = w_cmax + 128 * H;      // H
  float* w_q     = w_fq + H;              // H
  float* w_f2e   = w_q + H;               // MR (logits -> probs)
  float* w_part  = w_f2e + MR;            // 16*H
  float* w_ep    = w_part + 16 * H;       // H
  float* w_wv    = w_ep + H;              // H
  float* w_em    = w_wv + H;              // 256
  float* w_es    = w_em + 256;            // 256
  float* w_eacc  = w_es + 256;            // 256*H

  k1_colmax_part<<<dim3(3, 128), 256, 0, stream>>>(fact, w_cmax);
  k2_colmax_red <<<3, 256, 0, stream>>>(w_cmax, w_fq);
  k3_q          <<<3, 256, 0, stream>>>(w_fq, W, w_q);
  k4_f2e_logits <<<32, 256, 0, stream>>>(ep, w_q, w_f2e);
  k4b_softmax   <<<1, 256, 0, stream>>>(w_f2e);
  k4c_ep_partial<<<16, 256, 0, stream>>>(ep, w_f2e, w_part);
  k4d_ep_combine<<<3, 256, 0, stream>>>(w_part, w_ep, out);
  k4e_wv        <<<3, 256, 0, stream>>>(W, w_ep, w_wv);
  k5_e2f_partial<<<256, 256, 0, stream>>>(fact, w_wv, w_em, w_es, w_eacc);
  k6_combine    <<<3, 256, 0, stream>>>(w_em, w_es, w_eacc, out);
}